// TSPN_25194278158457
// MI455X (gfx1250) — compile-verified
//
#include <hip/hip_runtime.h>
#include <math.h>

// ---------------------------------------------------------------------------
// TSPN segment-stat kernel for MI455X (gfx1250).
//  - edge indices double-buffer prefetched to LDS via global_load_async_to_lds_b128
//    (CDNA5 async path, tracked with ASYNCcnt / s_wait_asynccnt)
//  - src attributes packed to float4, dst attributes to float2 -> 2 vector
//    gathers per edge instead of 6 scalar gathers (all L2-resident: 192MB L2)
//  - scatter via hardware atomics (u32 for degree, f32 fadd for sums)
//  NOTE: tspn_edges is defined FIRST so the disasm snippet shows it.
// ---------------------------------------------------------------------------

#define TSPN_TPB   256
#define TSPN_EPT   4                   // edges per thread per tile (one b128)
#define TSPN_TILE  (TSPN_TPB * TSPN_EPT)
#define TSPN_NLAY  6

static __device__ __forceinline__ void tspn_wait_async_le2() {
#if __has_builtin(__builtin_amdgcn_s_wait_asynccnt)
  __builtin_amdgcn_s_wait_asynccnt(2);
#else
  asm volatile("s_wait_asynccnt 2" ::: "memory");
#endif
}
static __device__ __forceinline__ void tspn_wait_async_0() {
#if __has_builtin(__builtin_amdgcn_s_wait_asynccnt)
  __builtin_amdgcn_s_wait_asynccnt(0);
#else
  asm volatile("s_wait_asynccnt 0" ::: "memory");
#endif
}

// Per-lane async copy of 16B from global to LDS. lds_addr is the raw LDS byte
// offset (low 32 bits of the generic pointer), gptr a per-lane global address.
static __device__ __forceinline__ void tspn_async_b128(unsigned lds_addr,
                                                       const void* gptr) {
  asm volatile("global_load_async_to_lds_b128 %0, %1, off"
               :: "v"(lds_addr), "v"(gptr)
               : "memory");
}

__global__ void __launch_bounds__(TSPN_TPB)
tspn_edges(const int* __restrict__ src, const int* __restrict__ dst,
           const float4* __restrict__ f4src, const float2* __restrict__ f2dst,
           float* __restrict__ sums, float* __restrict__ sqs,
           unsigned* __restrict__ deg, int n_edges) {
  __shared__ __align__(16) int sbuf[2][TSPN_TILE];
  __shared__ __align__(16) int dbuf[2][TSPN_TILE];

  const int tid    = threadIdx.x;
  const int nFull  = n_edges / TSPN_TILE;
  const int stride = (int)gridDim.x;

  const float kPi    = 3.14159265358979323846f;
  const float kTwoPi = 6.28318530717958647692f;

  unsigned lds_s[2], lds_d[2];
  lds_s[0] = (unsigned)(size_t)&sbuf[0][tid * TSPN_EPT];
  lds_s[1] = (unsigned)(size_t)&sbuf[1][tid * TSPN_EPT];
  lds_d[0] = (unsigned)(size_t)&dbuf[0][tid * TSPN_EPT];
  lds_d[1] = (unsigned)(size_t)&dbuf[1][tid * TSPN_EPT];

  auto issue = [&](int t, int b) {
    const int* gs = src + (size_t)t * TSPN_TILE + tid * TSPN_EPT;
    const int* gd = dst + (size_t)t * TSPN_TILE + tid * TSPN_EPT;
    tspn_async_b128(lds_s[b], (const void*)gs);
    tspn_async_b128(lds_d[b], (const void*)gd);
  };

  auto process = [&](int s, int d) {
    float4 sv = f4src[s];           // (eta, phi, energy, layer-bits): b128 gather
    float2 dv = f2dst[d];           // (eta, phi): b64 gather
    float deta = sv.x - dv.x;
    float dphi = sv.y - dv.y;
    dphi = (dphi >  kPi) ? dphi - kTwoPi : dphi;
    dphi = (dphi < -kPi) ? dphi + kTwoPi : dphi;
    (void)__hip_atomic_fetch_add(&deg[d], 1u, __ATOMIC_RELAXED,
                                 __HIP_MEMORY_SCOPE_AGENT);
    float r2 = deta * deta + dphi * dphi;
    if (sqrtf(r2) < 0.4f) {         // zero contribution otherwise -> skip exact
      float e = sv.z;
      int   l = __float_as_int(sv.w);
      int idx = d * TSPN_NLAY + l;
      (void)__hip_atomic_fetch_add(&sums[idx], e, __ATOMIC_RELAXED,
                                   __HIP_MEMORY_SCOPE_AGENT);
      (void)__hip_atomic_fetch_add(&sqs[idx], e * e, __ATOMIC_RELAXED,
                                   __HIP_MEMORY_SCOPE_AGENT);
    }
  };

  int t   = (int)blockIdx.x;
  int buf = 0;
  if (t < nFull) issue(t, 0);       // prime the pipeline (ASYNCcnt = 2)
  for (; t < nFull; t += stride) {
    int  nxt  = t + stride;
    bool pref = (nxt < nFull);
    if (pref) issue(nxt, buf ^ 1);  // ASYNCcnt = 4
    if (pref) tspn_wait_async_le2();   // current tile's 2 copies are done
    else      tspn_wait_async_0();     // drain
    // Each lane consumes only the 4 edges it staged -> no barrier needed.
    int4 sv4 = *reinterpret_cast<const int4*>(&sbuf[buf][tid * TSPN_EPT]);
    int4 dv4 = *reinterpret_cast<const int4*>(&dbuf[buf][tid * TSPN_EPT]);
    process(sv4.x, dv4.x);
    process(sv4.y, dv4.y);
    process(sv4.z, dv4.z);
    process(sv4.w, dv4.w);
    buf ^= 1;
  }

  // Tail (n_edges % TILE edges) via direct guarded loads.
  int g    = (int)blockIdx.x * TSPN_TPB + tid;
  int base = nFull * TSPN_TILE;
  int nt   = n_edges - base;
  if (g < nt) process(src[base + g], dst[base + g]);
}

__global__ void __launch_bounds__(TSPN_TPB)
tspn_zero(float* __restrict__ p, int n) {
  int i = blockIdx.x * TSPN_TPB + threadIdx.x;
  if (i < n) p[i] = 0.0f;
}

__global__ void __launch_bounds__(TSPN_TPB)
tspn_pack_src(const float* __restrict__ eta, const float* __restrict__ phi,
              const float* __restrict__ en, const int* __restrict__ lay,
              float4* __restrict__ out, int n) {
  int i = blockIdx.x * TSPN_TPB + threadIdx.x;
  if (i < n) out[i] = make_float4(eta[i], phi[i], en[i], __int_as_float(lay[i]));
}

__global__ void __launch_bounds__(TSPN_TPB)
tspn_pack_dst(const float* __restrict__ eta, const float* __restrict__ phi,
              float2* __restrict__ out, int n) {
  int i = blockIdx.x * TSPN_TPB + threadIdx.x;
  if (i < n) out[i] = make_float2(eta[i], phi[i]);
}

__global__ void __launch_bounds__(TSPN_TPB)
tspn_finalize(const float* __restrict__ sums, const float* __restrict__ sqs,
              const unsigned* __restrict__ deg, float* __restrict__ out,
              int n_dst) {
  int d = blockIdx.x * TSPN_TPB + threadIdx.x;
  if (d >= n_dst) return;
  float degf  = (float)deg[d];
  float dmax  = fmaxf(degf, 1.0f);
  float d1max = fmaxf(degf - 1.0f, 1.0f);
  bool  valid = degf > 1.0f;
#pragma unroll
  for (int j = 0; j < TSPN_NLAY; ++j) {
    float s    = sums[d * TSPN_NLAY + j];
    float q    = sqs[d * TSPN_NLAY + j];
    float mean = s / dmax;
    float var  = (q - degf * mean * mean) / d1max;
    var = fmaxf(var, 0.0f);
    float stdv = (var > 0.0f) ? sqrtf(var) : 0.0f;
    if (!valid) stdv = 0.0f;
    out[(size_t)d * 12 + j]     = s;
    out[(size_t)d * 12 + 6 + j] = stdv;
  }
}

extern "C" void kernel_launch(void* const* d_in, const int* in_sizes, int n_in,
                              void* d_out, int out_size, void* d_ws,
                              size_t ws_size, hipStream_t stream) {
  const int*   src     = (const int*)d_in[0];
  const int*   dst     = (const int*)d_in[1];
  const float* energy  = (const float*)d_in[2];
  const int*   layer   = (const int*)d_in[3];
  const float* eta_src = (const float*)d_in[4];
  const float* phi_src = (const float*)d_in[5];
  const float* eta_dst = (const float*)d_in[6];
  const float* phi_dst = (const float*)d_in[7];

  const int n_edges = in_sizes[0];
  const int n_src   = in_sizes[2];
  const int n_dst   = in_sizes[6];

  // Workspace layout (all 16B aligned):
  //   [sums: n_dst*6 f32][sqs: n_dst*6 f32][deg: n_dst u32]
  //   [f4src: n_src float4][f2dst: n_dst float2]
  char*     ws    = (char*)d_ws;
  float*    sums  = (float*)ws;
  float*    sqs   = sums + (size_t)n_dst * TSPN_NLAY;
  unsigned* deg   = (unsigned*)(sqs + (size_t)n_dst * TSPN_NLAY);
  size_t    off   = (((size_t)n_dst * 13 * 4) + 15) & ~(size_t)15;
  float4*   f4src = (float4*)(ws + off);
  float2*   f2dst = (float2*)(ws + off + (size_t)n_src * sizeof(float4));

  const int accN = n_dst * 13;  // zero writes 0x0 bits: valid for u32 deg too

  hipLaunchKernelGGL(tspn_zero, dim3((accN + TSPN_TPB - 1) / TSPN_TPB),
                     dim3(TSPN_TPB), 0, stream, (float*)ws, accN);
  hipLaunchKernelGGL(tspn_pack_src, dim3((n_src + TSPN_TPB - 1) / TSPN_TPB),
                     dim3(TSPN_TPB), 0, stream, eta_src, phi_src, energy, layer,
                     f4src, n_src);
  hipLaunchKernelGGL(tspn_pack_dst, dim3((n_dst + TSPN_TPB - 1) / TSPN_TPB),
                     dim3(TSPN_TPB), 0, stream, eta_dst, phi_dst, f2dst, n_dst);
  hipLaunchKernelGGL(tspn_edges, dim3(1024), dim3(TSPN_TPB), 0, stream, src,
                     dst, f4src, f2dst, sums, sqs, deg, n_edges);
  hipLaunchKernelGGL(tspn_finalize, dim3((n_dst + TSPN_TPB - 1) / TSPN_TPB),
                     dim3(TSPN_TPB), 0, stream, sums, sqs, deg, (float*)d_out,
                     n_dst);
}